// RSHxyz_81664508166970
// MI455X (gfx1250) — compile-verified
//
#include <hip/hip_runtime.h>

// ---------------------------------------------------------------------------
// Real spherical harmonics (max_l = 6) as   out[N,49] = mono[N,84] * W[84,49]
// using the CDNA5 full-precision f32 WMMA path: V_WMMA_F32_16X16X4_F32.
//   - 84 monomials of total degree <= 6  ->  21 K-slices of 4
//   - 49 channels padded to 64           ->  4 N-tiles of 16
// One wave32 handles a 16-point tile; 84 WMMAs per tile; W lives in LDS.
// ---------------------------------------------------------------------------

typedef float v2f __attribute__((ext_vector_type(2)));
typedef float v8f __attribute__((ext_vector_type(8)));

#define MAXL   6
#define KDIM   84    // # monomials of total degree <= MAXL
#define NSLICE 21    // KDIM / 4 (K per WMMA is 4)
#define NCH    49    // (MAXL+1)^2 output channels
#define NPAD   64    // channels padded to 4 WMMA n-tiles
#define BROW   72    // LDS row stride: 2*72 mod 64 == 16 -> lane halves hit
                     // disjoint bank groups when fetching B fragments

// Compile-time monomial enumeration: for l=0..6, a=0..l, b=0..l-a -> (a,b,l-a-b)
struct MonoTab { int px[KDIM]; int py[KDIM]; int pz[KDIM]; };
static constexpr MonoTab makeTab() {
  MonoTab t{};
  int k = 0;
  for (int l = 0; l <= MAXL; ++l)
    for (int a = 0; a <= l; ++a)
      for (int b = 0; b <= l - a; ++b) { t.px[k]=a; t.py[k]=b; t.pz[k]=l-a-b; ++k; }
  return t;
}
static constexpr MonoTab MT = makeTab();

// ---------------------------------------------------------------------------
// Setup: fold clm + scatter(dst) + ns scale into W[84][64].
// One thread per (k,j), scanning all T terms in fixed order -> deterministic.
// ---------------------------------------------------------------------------
__global__ void rsh_build_W(const float* __restrict__ clm,
                            const float* __restrict__ ns,
                            const int*   __restrict__ dst,
                            const int*   __restrict__ xp,
                            const int*   __restrict__ yp,
                            const int*   __restrict__ zp,
                            int T, float* __restrict__ Wg) {
  int idx = blockIdx.x * blockDim.x + threadIdx.x;
  if (idx >= KDIM * NPAD) return;
  int k = idx >> 6;          // monomial row
  int j = idx & 63;          // output channel
  float acc = 0.f;
  for (int t = 0; t < T; ++t) {
    int a = xp[t], b = yp[t], c = zp[t];
    int l = a + b + c;
    // same enumeration as MonoTab: offset(l) + a*(l+1) - a(a-1)/2 + b
    int kk = (l*(l+1)*(l+2))/6 + a*(l+1) - (a*(a-1))/2 + b;
    if (kk == k && dst[t] == j) acc += clm[t];
  }
  Wg[idx] = (j < NCH) ? acc * ns[j] : 0.f;
}

// ---------------------------------------------------------------------------
// Main kernel helpers: template-parameter indices keep all register arrays
// statically indexed (SROA-safe, everything stays in VGPRs).
// ---------------------------------------------------------------------------
template <int K>
__device__ __forceinline__ float mono(const float (&tx)[MAXL+1],
                                      const float (&ty)[MAXL+1],
                                      const float (&tz)[MAXL+1]) {
  return tx[MT.px[K]] * ty[MT.py[K]] * tz[MT.pz[K]];
}

template <int I>
__device__ __forceinline__ void kslice(const float (&tx)[MAXL+1],
                                       const float (&ty)[MAXL+1],
                                       const float (&tz)[MAXL+1],
                                       bool hiHalf, int laneN,
                                       const float* __restrict__ Wl,
                                       v8f& a0, v8f& a1, v8f& a2, v8f& a3) {
  // All 4 monomials of this K-slice; each lane keeps the pair its half owns.
  // A 16x4 f32 layout: lanes 0-15 hold (K0,K1), lanes 16-31 hold (K2,K3).
  float m0 = mono<4*I+0>(tx, ty, tz);
  float m1 = mono<4*I+1>(tx, ty, tz);
  float m2 = mono<4*I+2>(tx, ty, tz);
  float m3 = mono<4*I+3>(tx, ty, tz);
  v2f a;
  a.x = hiHalf ? m2 : m0;
  a.y = hiHalf ? m3 : m1;
  const int rbase = 4*I + (hiHalf ? 2 : 0);
  const float* w0 = Wl + rbase * BROW + laneN;       // B row K (this half)
  const float* w1 = w0 + BROW;                       // B row K+1
  {
    v2f b; b.x = w0[0];  b.y = w1[0];
    a0 = __builtin_amdgcn_wmma_f32_16x16x4_f32(false, a, false, b, (short)0, a0, false, false);
  }
  {
    v2f b; b.x = w0[16]; b.y = w1[16];
    a1 = __builtin_amdgcn_wmma_f32_16x16x4_f32(false, a, false, b, (short)0, a1, false, false);
  }
  {
    v2f b; b.x = w0[32]; b.y = w1[32];
    a2 = __builtin_amdgcn_wmma_f32_16x16x4_f32(false, a, false, b, (short)0, a2, false, false);
  }
  {
    v2f b; b.x = w0[48]; b.y = w1[48];
    a3 = __builtin_amdgcn_wmma_f32_16x16x4_f32(false, a, false, b, (short)0, a3, false, false);
  }
}

__global__ __launch_bounds__(256)
void rsh_wmma_kernel(const float* __restrict__ xyz,
                     const float* __restrict__ Wg,
                     float* __restrict__ out,
                     int npts, int ntiles) {
  __shared__ float Wl[KDIM * BROW];
  for (int i = threadIdx.x; i < KDIM * NPAD; i += blockDim.x) {
    int r = i >> 6, c = i & 63;
    Wl[r * BROW + c] = Wg[i];
  }
  __syncthreads();

  const int  lane    = threadIdx.x & 31;
  const int  wave    = threadIdx.x >> 5;
  const int  laneN   = lane & 15;
  const bool hiHalf  = lane >= 16;
  const int  wavesAll = gridDim.x * (blockDim.x >> 5);

  for (int tile = blockIdx.x * (blockDim.x >> 5) + wave; tile < ntiles;
       tile += wavesAll) {
    // Lane L and L+16 both own point (tile*16 + L%16) per the A layout.
    int p  = tile * 16 + laneN;
    int pc = p < npts ? p : npts - 1;        // clamp loads; EXEC stays all-1s
    float x = xyz[3*pc + 0];
    float y = xyz[3*pc + 1];
    float z = xyz[3*pc + 2];

    // Power tables, literal indices only -> pure VGPRs.
    float tx[MAXL+1], ty[MAXL+1], tz[MAXL+1];
    tx[0]=1.f;      ty[0]=1.f;      tz[0]=1.f;
    tx[1]=x;        ty[1]=y;        tz[1]=z;
    tx[2]=tx[1]*x;  ty[2]=ty[1]*y;  tz[2]=tz[1]*z;
    tx[3]=tx[2]*x;  ty[3]=ty[2]*y;  tz[3]=tz[2]*z;
    tx[4]=tx[3]*x;  ty[4]=ty[3]*y;  tz[4]=tz[3]*z;
    tx[5]=tx[4]*x;  ty[5]=ty[4]*y;  tz[5]=tz[4]*z;
    tx[6]=tx[5]*x;  ty[6]=ty[5]*y;  tz[6]=tz[5]*z;

    v8f a0 = {}, a1 = {}, a2 = {}, a3 = {};

    kslice< 0>(tx,ty,tz,hiHalf,laneN,Wl,a0,a1,a2,a3);
    kslice< 1>(tx,ty,tz,hiHalf,laneN,Wl,a0,a1,a2,a3);
    kslice< 2>(tx,ty,tz,hiHalf,laneN,Wl,a0,a1,a2,a3);
    kslice< 3>(tx,ty,tz,hiHalf,laneN,Wl,a0,a1,a2,a3);
    kslice< 4>(tx,ty,tz,hiHalf,laneN,Wl,a0,a1,a2,a3);
    kslice< 5>(tx,ty,tz,hiHalf,laneN,Wl,a0,a1,a2,a3);
    kslice< 6>(tx,ty,tz,hiHalf,laneN,Wl,a0,a1,a2,a3);
    kslice< 7>(tx,ty,tz,hiHalf,laneN,Wl,a0,a1,a2,a3);
    kslice< 8>(tx,ty,tz,hiHalf,laneN,Wl,a0,a1,a2,a3);
    kslice< 9>(tx,ty,tz,hiHalf,laneN,Wl,a0,a1,a2,a3);
    kslice<10>(tx,ty,tz,hiHalf,laneN,Wl,a0,a1,a2,a3);
    kslice<11>(tx,ty,tz,hiHalf,laneN,Wl,a0,a1,a2,a3);
    kslice<12>(tx,ty,tz,hiHalf,laneN,Wl,a0,a1,a2,a3);
    kslice<13>(tx,ty,tz,hiHalf,laneN,Wl,a0,a1,a2,a3);
    kslice<14>(tx,ty,tz,hiHalf,laneN,Wl,a0,a1,a2,a3);
    kslice<15>(tx,ty,tz,hiHalf,laneN,Wl,a0,a1,a2,a3);
    kslice<16>(tx,ty,tz,hiHalf,laneN,Wl,a0,a1,a2,a3);
    kslice<17>(tx,ty,tz,hiHalf,laneN,Wl,a0,a1,a2,a3);
    kslice<18>(tx,ty,tz,hiHalf,laneN,Wl,a0,a1,a2,a3);
    kslice<19>(tx,ty,tz,hiHalf,laneN,Wl,a0,a1,a2,a3);
    kslice<20>(tx,ty,tz,hiHalf,laneN,Wl,a0,a1,a2,a3);

    // D layout: VGPR v -> row (v + 8*hiHalf), col laneN (within each n-tile).
    const int prow = tile * 16 + (hiHalf ? 8 : 0);
#pragma unroll
    for (int v = 0; v < 8; ++v) {
      int pt = prow + v;
      if (pt < npts) {
        float* orow = out + (size_t)pt * NCH;
        orow[laneN]      = a0[v];
        orow[16 + laneN] = a1[v];
        orow[32 + laneN] = a2[v];
        if (laneN == 0) orow[48] = a3[v];   // only channel 48 of n-tile 3 is real
      }
    }
  }
}

extern "C" void kernel_launch(void* const* d_in, const int* in_sizes, int n_in,
                              void* d_out, int out_size, void* d_ws, size_t ws_size,
                              hipStream_t stream) {
  (void)n_in; (void)out_size; (void)ws_size;
  const float* xyz = (const float*)d_in[0];
  const float* clm = (const float*)d_in[1];
  const float* ns  = (const float*)d_in[2];
  const int*   dst = (const int*)d_in[3];
  const int*   xp  = (const int*)d_in[4];
  const int*   yp  = (const int*)d_in[5];
  const int*   zp  = (const int*)d_in[6];
  float* out = (float*)d_out;
  float* Wg  = (float*)d_ws;              // KDIM*NPAD floats = 21504 B

  const int npts = in_sizes[0] / 3;
  const int T    = in_sizes[1];

  const int wThreads = KDIM * NPAD;
  rsh_build_W<<<(wThreads + 255) / 256, 256, 0, stream>>>(clm, ns, dst, xp, yp, zp, T, Wg);

  const int ntiles = (npts + 15) / 16;
  const int wavesPerBlock = 8;            // 256 threads = 8 wave32
  const int blocks = (ntiles + wavesPerBlock - 1) / wavesPerBlock;
  rsh_wmma_kernel<<<blocks, 32 * wavesPerBlock, 0, stream>>>(xyz, Wg, out, npts, ntiles);
}